// NPairLoss_59605556134109
// MI455X (gfx1250) — compile-verified
//
#include <hip/hip_runtime.h>

// N-pair loss on MI455X (gfx1250, wave32).
// sim = E * E^T via V_WMMA_F32_16X16X4_F32 (full f32 precision),
// fused exp/where epilogue, deterministic two-stage reduction.

#define NROWS   8192
#define DIMK    128
#define MARGIN  0.1f
#define WG_ROWS 64     // rows per workgroup (4 row-tiles of 16)
#define CT      32     // columns per iteration (2 col-tiles of 16)
#define LDP     132    // padded LDS row stride in floats (bank-conflict-free, 16B aligned)

typedef __attribute__((ext_vector_type(2))) float v2f;
typedef __attribute__((ext_vector_type(8))) float v8f;

__global__ __launch_bounds__(256)
void npair_main_kernel(const float* __restrict__ emb,
                       const int*   __restrict__ labels,
                       float*       __restrict__ blocksum) {
    __shared__ float Alds[WG_ROWS * LDP];
    __shared__ float Blds[CT * LDP];
    __shared__ int   lblA[WG_ROWS];
    __shared__ int   lblB[CT];
    __shared__ float rowpart[WG_ROWS][2];
    __shared__ float logs[WG_ROWS];

    const int tid   = threadIdx.x;
    const int rbase = blockIdx.x * WG_ROWS;

    // ---- stage A block (64 x 128 f32) into LDS, coalesced float4 ----
    #pragma unroll
    for (int i = 0; i < 8; ++i) {
        int idx = tid + i * 256;          // 0..2047 float4 slots
        int r   = idx >> 5;               // row 0..63
        int cg  = idx & 31;               // float4 group 0..31
        float4 v = ((const float4*)(emb + (size_t)(rbase + r) * DIMK))[cg];
        *(float4*)(Alds + r * LDP + cg * 4) = v;
    }
    if (tid < WG_ROWS) lblA[tid] = labels[rbase + tid];
    __syncthreads();

    const int wave  = tid >> 5;
    const int lane  = tid & 31;
    const int l16   = lane & 15;
    const int hi    = lane >> 4;          // lane half: selects K pair / M half
    const int rtile = wave >> 1;          // 0..3
    const int ctile = wave & 1;           // 0..1

    // ---- hoist this wave's A fragments into registers (32 K-steps) ----
    // A f32 16x4 layout: lanes 0-15 hold {K=0,K=1}, lanes 16-31 hold {K=2,K=3}
    v2f areg[32];
    {
        const float* ap = Alds + (rtile * 16 + l16) * LDP + 2 * hi;
        #pragma unroll
        for (int kk = 0; kk < 32; ++kk)
            areg[kk] = *(const v2f*)(ap + kk * 4);
    }
    int la[8];
    #pragma unroll
    for (int r = 0; r < 8; ++r) la[r] = lblA[rtile * 16 + hi * 8 + r];

    float rowacc[8];
    #pragma unroll
    for (int r = 0; r < 8; ++r) rowacc[r] = 0.0f;

    // ---- sweep all 8192 columns in tiles of 32 ----
    for (int c0 = 0; c0 < NROWS; c0 += CT) {
        __syncthreads();
        // stage B block (32 x 128 f32) into LDS
        #pragma unroll
        for (int i = 0; i < 4; ++i) {
            int idx = tid + i * 256;      // 0..1023 float4 slots
            int r   = idx >> 5;
            int cg  = idx & 31;
            float4 v = ((const float4*)(emb + (size_t)(c0 + r) * DIMK))[cg];
            *(float4*)(Blds + r * LDP + cg * 4) = v;
        }
        if (tid < CT) lblB[tid] = labels[c0 + tid];
        if (c0 + CT < NROWS)  // prefetch next column tile (global_prefetch_b8)
            __builtin_prefetch(emb + (size_t)(c0 + CT) * DIMK + tid * 16, 0, 0);
        __syncthreads();

        // ---- 16x16 f32 Gram tile: 32 x V_WMMA_F32_16X16X4_F32 ----
        v8f c = {0.f, 0.f, 0.f, 0.f, 0.f, 0.f, 0.f, 0.f};
        const float* bp = Blds + (ctile * 16 + l16) * LDP + 2 * hi;
        #pragma unroll
        for (int kk = 0; kk < 32; ++kk) {
            v2f b = *(const v2f*)(bp + kk * 4);
            c = __builtin_amdgcn_wmma_f32_16x16x4_f32(
                    false, areg[kk], false, b, (short)0, c, false, false);
        }

        // ---- fused epilogue: exp(eq ? M-s : s-M), accumulate per row ----
        const int lb = lblB[ctile * 16 + l16];
        #pragma unroll
        for (int r = 0; r < 8; ++r) {
            float s   = c[r];
            float arg = (la[r] == lb) ? (MARGIN - s) : (s - MARGIN);
            rowacc[r] += __expf(arg);
        }
    }

    // ---- reduce 16 lanes (per wave32 half) per row, deterministic ----
    #pragma unroll
    for (int r = 0; r < 8; ++r) {
        float v = rowacc[r];
        v += __shfl_xor(v, 1, 32);
        v += __shfl_xor(v, 2, 32);
        v += __shfl_xor(v, 4, 32);
        v += __shfl_xor(v, 8, 32);
        rowacc[r] = v;
    }
    if (l16 == 0) {
        #pragma unroll
        for (int r = 0; r < 8; ++r)
            rowpart[rtile * 16 + hi * 8 + r][ctile] = rowacc[r];
    }
    __syncthreads();

    if (tid < WG_ROWS)
        logs[tid] = logf(rowpart[tid][0] + rowpart[tid][1]);
    __syncthreads();

    if (tid == 0) {
        float s = 0.0f;
        for (int i = 0; i < WG_ROWS; ++i) s += logs[i];
        blocksum[blockIdx.x] = s;
    }
}

__global__ void npair_final_kernel(const float* __restrict__ blocksum,
                                   float* __restrict__ out) {
    if (threadIdx.x == 0 && blockIdx.x == 0) {
        float s = 0.0f;
        for (int i = 0; i < NROWS / WG_ROWS; ++i) s += blocksum[i];
        out[0] = s / (float)NROWS;
    }
}

extern "C" void kernel_launch(void* const* d_in, const int* in_sizes, int n_in,
                              void* d_out, int out_size, void* d_ws, size_t ws_size,
                              hipStream_t stream) {
    (void)in_sizes; (void)n_in; (void)out_size; (void)ws_size;
    const float* emb    = (const float*)d_in[0];
    const int*   labels = (const int*)d_in[1];
    float*       out    = (float*)d_out;
    float*       bsum   = (float*)d_ws;   // 128 floats of scratch

    npair_main_kernel<<<NROWS / WG_ROWS, 256, 0, stream>>>(emb, labels, bsum);
    npair_final_kernel<<<1, 32, 0, stream>>>(bsum, out);
}